// CDFNormalizer_67156108640724
// MI455X (gfx1250) — compile-verified
//
#include <hip/hip_runtime.h>
#include <stdint.h>

// CDF normalizer: idx = searchsorted(quantiles[:,d], x[n,d]); z = erfinv(2*clip(idx/1023)-1)*sqrt2
// Memory-bound streaming kernel (512MB @ 23.3TB/s ~ 22us floor).
// - quantile table (128KB) staged to LDS with gfx1250 async global->LDS DMA (ASYNCcnt),
//   32 back-to-back DMAs using the 24-bit INST_OFFSET (applies to both LDS+global sides)
// - [BINS][32] LDS layout => bank-conflict-free binary search (lane == feature dim)
// - erfinv folded into a 1024-entry per-block LDS LUT (result depends only on idx)
// - 10-step branchless lower_bound: since clip() maps idx=1023 and idx=1024 to the same
//   output, min(count,1023) is sufficient -> the 11th probe is dropped entirely.

#define BINS 1024
#define DIMS 32
#define LUTN 1024   // idx 1023 and 1024 collapse under clip(); 1024 entries suffice
#define TPB  256

__device__ __forceinline__ float erfinvf_approx(float x) {
    // Giles (2010) single-precision erfinv
    float w = -logf((1.0f - x) * (1.0f + x));
    float p;
    if (w < 5.0f) {
        w -= 2.5f;
        p = 2.81022636e-08f;
        p = fmaf(p, w, 3.43273939e-07f);
        p = fmaf(p, w, -3.5233877e-06f);
        p = fmaf(p, w, -4.39150654e-06f);
        p = fmaf(p, w, 0.00021858087f);
        p = fmaf(p, w, -0.00125372503f);
        p = fmaf(p, w, -0.00417768164f);
        p = fmaf(p, w, 0.246640727f);
        p = fmaf(p, w, 1.50140941f);
    } else {
        w = sqrtf(w) - 3.0f;
        p = -0.000200214257f;
        p = fmaf(p, w, 0.000100950558f);
        p = fmaf(p, w, 0.00134934322f);
        p = fmaf(p, w, -0.00367342844f);
        p = fmaf(p, w, 0.00573950773f);
        p = fmaf(p, w, -0.0076224613f);
        p = fmaf(p, w, 0.00943887047f);
        p = fmaf(p, w, 1.00167406f);
        p = fmaf(p, w, 2.83297682f);
    }
    return p * x;
}

// Branchless lower_bound over 1024 sorted entries (column d of [bin][dim] LDS table).
// Returns posf = 32 * min(#entries < v, 1023)  (float-index, pre-scaled by DIMS).
// Per-step probe offset (step-1)*32 floats is a compile-time constant that folds into
// the ds_load immediate offset field ((step-1)*128 bytes <= 65408 fits the 16-bit imm).
// Probe index pos+step-1 <= 1022 for all 10 steps -> always in bounds, no clamp.
__device__ __forceinline__ int search_col(const float* __restrict__ qcol, float v) {
    int posf = 0;
#pragma unroll
    for (int step = 512; step >= 1; step >>= 1) {
        float qv = qcol[posf + (step - 1) * DIMS];
        posf += (qv < v) ? (step * DIMS) : 0;
    }
    return posf;   // caller indexes zlut[posf >> 5]
}

__global__ __launch_bounds__(TPB) void cdf_normalizer_kernel(
    const float* __restrict__ x,
    const float* __restrict__ q,     // [BINS][DIMS], sorted along bins per dim
    float* __restrict__ out,
    long long total_ll)              // N * DIMS elements
{
    __shared__ float qs[BINS * DIMS];   // 128 KB quantile table, [bin][dim]
    __shared__ float zlut[LUTN];        // 4 KB: idx -> erfinv(2*clip(idx/1023)-1)*sqrt2

    const int tid = threadIdx.x;

    // ---- Stage quantile table: 32 back-to-back async global->LDS DMAs ----
    {
        const uint32_t la = (uint32_t)(uintptr_t)(&qs[0]) + (uint32_t)tid * 16u;
        const uint64_t ga = (uint64_t)(uintptr_t)q + (uint64_t)tid * 16u;
#pragma unroll
        for (int k = 0; k < (BINS * DIMS * 4) / (16 * TPB); ++k) {   // 32 iters
            asm volatile("global_load_async_to_lds_b128 %0, %1, off offset:%c2"
                         :
                         : "v"(la), "v"(ga), "i"(k * 16 * TPB)
                         : "memory");
        }
    }

    // ---- Build the erfinv LUT while the DMA is in flight (independent work) ----
    for (int i = tid; i < LUTN; i += TPB) {
        float u = (float)i * (1.0f / (float)(BINS - 1));
        u = fminf(fmaxf(u, 1e-6f), 1.0f - 1e-6f);
        zlut[i] = erfinvf_approx(2.0f * u - 1.0f) * 1.41421356f;
    }

    asm volatile("s_wait_asynccnt 0" ::: "memory");  // this wave's DMA done
    __syncthreads();                                 // everyone's DMA + LUT visible

    // ---- Streaming pass: warp handles 128-element chunks (4 rows of 32 dims) ----
    const unsigned total = (unsigned)total_ll;        // 2^26 elements: 32-bit indexing
    const unsigned lane  = (unsigned)tid & 31u;       // lane == feature dim d
    const float* qcol = &qs[lane];                    // column base, stride 32 floats
    const unsigned warp   = blockIdx.x * (TPB >> 5) + ((unsigned)tid >> 5);
    const unsigned nwarps = gridDim.x * (TPB >> 5);
    const unsigned nfull  = total >> 7;               // full 128-element chunks

    for (unsigned c = warp; c < nfull; c += nwarps) {
        unsigned base = c * 128u + lane;
        // 4 independent loads -> 4 independent search chains in flight
        float v0 = __builtin_nontemporal_load(&x[base]);
        float v1 = __builtin_nontemporal_load(&x[base + 32u]);
        float v2 = __builtin_nontemporal_load(&x[base + 64u]);
        float v3 = __builtin_nontemporal_load(&x[base + 96u]);
        int p0 = search_col(qcol, v0);
        int p1 = search_col(qcol, v1);
        int p2 = search_col(qcol, v2);
        int p3 = search_col(qcol, v3);
        __builtin_nontemporal_store(zlut[p0 >> 5], &out[base]);
        __builtin_nontemporal_store(zlut[p1 >> 5], &out[base + 32u]);
        __builtin_nontemporal_store(zlut[p2 >> 5], &out[base + 64u]);
        __builtin_nontemporal_store(zlut[p3 >> 5], &out[base + 96u]);
    }

    // ---- Tail: leftover elements (< 128), guarded ----
    const unsigned gtid = blockIdx.x * TPB + (unsigned)tid;
    for (unsigned e = nfull * 128u + gtid; e < total; e += gridDim.x * TPB) {
        float v = x[e];
        int posf = search_col(&qs[e & 31u], v);
        out[e] = zlut[posf >> 5];
    }
}

extern "C" void kernel_launch(void* const* d_in, const int* in_sizes, int n_in,
                              void* d_out, int out_size, void* d_ws, size_t ws_size,
                              hipStream_t stream) {
    const float* x = (const float*)d_in[0];   // [N, 32] fp32
    const float* q = (const float*)d_in[1];   // [1024, 32] fp32, sorted per column
    float* out = (float*)d_out;               // [N, 32] fp32
    (void)d_ws; (void)ws_size; (void)n_in; (void)in_sizes;

    long long total = (long long)out_size;
    int blocks = 512;                         // 2 blocks/WGP (132KB LDS each);
                                              // 4096 warps -> ~128 chunks per warp
    hipLaunchKernelGGL(cdf_normalizer_kernel, dim3(blocks), dim3(TPB), 0, stream,
                       x, q, out, total);
}